// LambdaRankLoss_36532991819997
// MI455X (gfx1250) — compile-verified
//
#include <hip/hip_runtime.h>
#include <hip/hip_bf16.h>

typedef float v2f __attribute__((ext_vector_type(2)));
typedef float v8f __attribute__((ext_vector_type(8)));

#define MAXG 8          // grades are 0..4; pad to 8 for safety
#define KMAX 16         // top-K storage (K == 10)
#define NBLK 8
#define NTHR 256

struct Hdr {
    int   ktop;
    float idcg;
    float inv_idcg;
    float n_pairs;
    float tp[KMAX];     // predictions of top-K (ideal-rank < K) items
    float tg[KMAX];     // gains 2^t - 1
    float td[KMAX];     // discounts 1/log2(rank+2)
    float tt[KMAX];     // target grade (float)
    int   tidx[KMAX];   // original index
    float partials[64]; // per-block partial sums
};

// ---------------------------------------------------------------------------
// Kernel 1: histogram of grades -> n_pairs; stable-rank scan -> top-K + IDCG.
// One block. targets staged to LDS so the thread-0 scan stays on-chip.
// ---------------------------------------------------------------------------
__global__ void lr_prep(const float* __restrict__ pred,
                        const float* __restrict__ tgt,
                        int n, int k, Hdr* __restrict__ h) {
    __shared__ float sh_t[8192];
    __shared__ int   cnt[MAXG];
    const int tid = threadIdx.x;
    if (tid < MAXG) cnt[tid] = 0;
    __syncthreads();
    for (int i = tid; i < n; i += blockDim.x) {
        float t = tgt[i];
        if (i < 8192) sh_t[i] = t;
        int g = (int)t;
        g = g < 0 ? 0 : (g > MAXG - 1 ? MAXG - 1 : g);
        atomicAdd(&cnt[g], 1);
    }
    __syncthreads();
    if (tid == 0) {
        // hi[g] = #items with grade > g  (start rank of grade-g run in ideal order)
        int hi[MAXG];
        int acc = 0;
        for (int g = MAXG - 1; g >= 0; --g) { hi[g] = acc; acc += cnt[g]; }
        // n_pairs from histogram (exact; total <= n^2/2 < 2^24 fits f32)
        double np = 0.0;
        for (int a = 1; a < MAXG; ++a)
            for (int b = 0; b < a; ++b)
                np += (double)cnt[a] * (double)cnt[b];
        // stable ranks: within a grade, ideal order == index order
        int seen[MAXG];
        for (int g = 0; g < MAXG; ++g) seen[g] = 0;
        for (int s = 0; s < KMAX; ++s) {
            h->tp[s] = 0.f; h->tg[s] = 0.f; h->td[s] = 0.f; h->tt[s] = 0.f;
            h->tidx[s] = -1;
        }
        int   kt   = 0;
        float idcg = 0.f;
        for (int i = 0; i < n; ++i) {
            if (kt >= k) break;                 // exactly k items have rank < k
            float t = (i < 8192) ? sh_t[i] : tgt[i];
            int g = (int)t;
            g = g < 0 ? 0 : (g > MAXG - 1 ? MAXG - 1 : g);
            int r = hi[g] + seen[g];
            seen[g]++;
            if (r < k) {
                float disc = 1.0f / log2f((float)(r + 2));
                float gain = exp2f(t) - 1.0f;
                h->tp[kt] = pred[i];
                h->tg[kt] = gain;
                h->td[kt] = disc;
                h->tt[kt] = t;
                h->tidx[kt] = i;
                idcg += gain * disc;
                kt++;
            }
        }
        h->ktop     = kt;
        h->idcg     = idcg;
        h->inv_idcg = (idcg > 0.f) ? 1.0f / idcg : 0.0f;
        h->n_pairs  = (float)np;
    }
}

// ---------------------------------------------------------------------------
// Kernel 2: for each item j, accumulate loss against the <=K top items whose
// target exceeds t_j. Wave reduction done with V_WMMA_F32_16X16X4_F32:
// D = A x ones + 0  => row-sums of the wave's 64 per-lane partials, with the
// total recoverable from lanes 0 and 16 (column 0 of each half).
// ---------------------------------------------------------------------------
__global__ void lr_pairs(const float* __restrict__ pred,
                         const float* __restrict__ tgt,
                         int n, Hdr* __restrict__ h, int tot_threads) {
    __shared__ float l_p[KMAX], l_g[KMAX], l_d[KMAX], l_t[KMAX];
    __shared__ int   l_i[KMAX];
    __shared__ int   s_kt;
    __shared__ float s_inv;
    __shared__ float wsum[16];   // 8 waves x 2 half-sums

    const int tid = threadIdx.x;
    if (tid < KMAX) {
        l_p[tid] = h->tp[tid];
        l_g[tid] = h->tg[tid];
        l_d[tid] = h->td[tid];
        l_t[tid] = h->tt[tid];
        l_i[tid] = h->tidx[tid];
    }
    if (tid == 0) { s_kt = h->ktop; s_inv = h->inv_idcg; }
    __syncthreads();

    const int   kt  = s_kt;
    const float inv = s_inv;
    const int gtid  = blockIdx.x * blockDim.x + tid;

    v2f a; a.x = 0.f; a.y = 0.f;   // this lane's two A-matrix slots
    int sl = 0;
    for (int j = gtid; j < n; j += tot_threads) {
        float tj = tgt[j];
        float pj = pred[j];
        float gj = exp2f(tj) - 1.0f;
        float dj = 0.f;                          // disc_j != 0 only if j is top-K
        for (int q = 0; q < kt; ++q)
            if (l_i[q] == j) dj = l_d[q];
        float lj = 0.f;
        for (int q = 0; q < kt; ++q) {
            if (l_t[q] > tj) {                   // mask: t_i > t_j
                float d   = l_p[q] - pj;         // SIGMA == 1
                float sgm = 1.0f / (1.0f + expf(-d));
                lj += -logf(sgm + 1e-10f) * (l_g[q] - gj) * (l_d[q] - dj);
            }
        }
        lj *= inv;
        if (sl == 0) a.x += lj; else a.y += lj;
        sl ^= 1;
    }

    // Wave-level 64 -> 16 reduction on the matrix pipe (EXEC is all-1s here).
    v2f ones; ones.x = 1.0f; ones.y = 1.0f;
    v8f c = {};
    v8f dacc = __builtin_amdgcn_wmma_f32_16x16x4_f32(
        /*neg_a=*/false, a, /*neg_b=*/false, ones,
        /*c_mod=*/(short)0, c, /*reuse_a=*/false, /*reuse_b=*/false);
    float s8 = dacc[0] + dacc[1] + dacc[2] + dacc[3] +
               dacc[4] + dacc[5] + dacc[6] + dacc[7];

    const int lane = tid & 31;
    const int wv   = tid >> 5;
    if (lane == 0)  wsum[2 * wv]     = s8;   // rows 0..7, column 0
    if (lane == 16) wsum[2 * wv + 1] = s8;   // rows 8..15, column 0
    __syncthreads();
    if (tid == 0) {
        float t = 0.f;
        const int nw = (int)(blockDim.x >> 5);
        for (int i = 0; i < 2 * nw; ++i) t += wsum[i];
        h->partials[blockIdx.x] = t;
    }
}

// ---------------------------------------------------------------------------
// Kernel 3: fixed-order final sum + normalization (deterministic).
// ---------------------------------------------------------------------------
__global__ void lr_final(const Hdr* __restrict__ h, float* __restrict__ out, int nb) {
    if (threadIdx.x == 0 && blockIdx.x == 0) {
        float t = 0.f;
        for (int i = 0; i < nb; ++i) t += h->partials[i];
        out[0] = (h->n_pairs > 0.f) ? (t / h->n_pairs) : 0.0f;
    }
}

extern "C" void kernel_launch(void* const* d_in, const int* in_sizes, int n_in,
                              void* d_out, int out_size, void* d_ws, size_t ws_size,
                              hipStream_t stream) {
    const float* pred = (const float*)d_in[0];
    const float* tgt  = (const float*)d_in[1];
    const int n = in_sizes[0];
    const int k = (n < 10) ? n : 10;
    Hdr* h = (Hdr*)d_ws;

    lr_prep <<<1, NTHR, 0, stream>>>(pred, tgt, n, k, h);
    lr_pairs<<<NBLK, NTHR, 0, stream>>>(pred, tgt, n, h, NBLK * NTHR);
    lr_final<<<1, 32, 0, stream>>>(h, (float*)d_out, NBLK);
}